// MMD_66408784331046
// MI455X (gfx1250) — compile-verified
//
#include <hip/hip_runtime.h>
#include <hip/hip_bf16.h>

typedef __attribute__((ext_vector_type(2))) float v2f;
typedef __attribute__((ext_vector_type(8))) float v8f;

#define DDIM 256
#define BROWS 4096
#define ROWS_PER_WAVE 64
#define WAVES_PER_BLOCK 8
#define ROWS_PER_BLOCK (ROWS_PER_WAVE * WAVES_PER_BLOCK)          // 512
#define NUM_BLOCKS (2 * 16 * (BROWS / ROWS_PER_BLOCK))            // 256

// Monotonic float<->uint order-preserving transform (for atomic max/min).
__device__ __forceinline__ unsigned flipf(float f) {
    unsigned u = __float_as_uint(f);
    return u ^ ((unsigned)((int)u >> 31) | 0x80000000u);
}
__device__ __forceinline__ float unflipf(unsigned u) {
    unsigned v = (u & 0x80000000u) ? (u ^ 0x80000000u) : ~u;
    return __uint_as_float(v);
}

// Workspace layout (floats unless noted):
//   ws[0..255]   : column sums of source (unscaled)
//   ws[256..511] : column sums of target (unscaled)
//   ws[512]      : sum of squares over both matrices (unscaled)
//   ws_u[513]    : flipped max of source   (init 0)
//   ws_u[514]    : flipped min of source   (init 0xFFFFFFFF)

__global__ __launch_bounds__(256) void mmd_zero(float* ws) {
    int i = threadIdx.x;
    ws[i] = 0.0f;
    ws[i + 256] = 0.0f;
    if (i == 0) {
        ws[512] = 0.0f;
        ((unsigned*)ws)[513] = 0u;
        ((unsigned*)ws)[514] = 0xFFFFFFFFu;
    }
}

__global__ __launch_bounds__(256) void mmd_reduce(const float* __restrict__ src,
                                                  const float* __restrict__ tgt,
                                                  float* __restrict__ ws) {
    const int tid  = threadIdx.x;
    const int lane = tid & 31;
    const int wave = tid >> 5;
    const int b    = blockIdx.x;

    const int mat    = b & 1;          // 0 = source, 1 = target
    const int stripe = (b >> 1) & 15;  // which 16-column stripe
    const int rowblk = b >> 5;         // which 512-row block

    const float* base = mat ? tgt : src;
    const int col0 = stripe * 16;
    const int row0 = rowblk * ROWS_PER_BLOCK + wave * ROWS_PER_WAVE;

    // B-operand lane addressing for v_wmma_f32_16x16x4_f32:
    // lanes 0..15 feed rows {k, k+2}, lanes 16..31 feed rows {k+1, k+3},
    // column = col0 + (lane & 15). With A == all-ones the accumulator row 0
    // holds the exact column sums regardless of the internal K-slot labeling.
    const int half = lane >> 4;        // 0 or 1
    const int n    = lane & 15;
    const float* p = base + (row0 + half) * DDIM + col0 + n;

    v2f a; a.x = 1.0f; a.y = 1.0f;     // 16x4 all-ones A matrix
    v8f c = {};
    float sumsq = 0.0f;
    float mx = -3.402823466e38f;
    float mn =  3.402823466e38f;

#pragma unroll
    for (int k = 0; k < ROWS_PER_WAVE; k += 4) {
        float bx = p[k * DDIM];
        float by = p[(k + 2) * DDIM];
        v2f bb; bb.x = bx; bb.y = by;
        // D = ones(16x4) * B(4x16) + C  -> per-column partial sums
        c = __builtin_amdgcn_wmma_f32_16x16x4_f32(
                /*neg_a=*/false, a, /*neg_b=*/false, bb,
                /*c_mod=*/(short)0, c, /*reuse_a=*/false, /*reuse_b=*/false);
        sumsq = fmaf(bx, bx, sumsq);
        sumsq = fmaf(by, by, sumsq);
        mx = fmaxf(mx, fmaxf(bx, by));
        mn = fminf(mn, fminf(bx, by));
    }

    // Accumulator VGPR0, lanes 0..15: M=0 row == column sums for this stripe.
    if (lane < 16) {
        atomicAdd(&ws[mat * DDIM + col0 + lane], c[0]);
    }

    // Wave-level reduction of sumsq / max / min.
    for (int off = 16; off; off >>= 1) {
        sumsq += __shfl_down(sumsq, off);
        mx = fmaxf(mx, __shfl_down(mx, off));
        mn = fminf(mn, __shfl_down(mn, off));
    }
    if (lane == 0) {
        atomicAdd(&ws[512], sumsq);
        if (mat == 0) {
            atomicMax(((unsigned*)ws) + 513, flipf(mx));
            atomicMin(((unsigned*)ws) + 514, flipf(mn));
        }
    }
}

__global__ __launch_bounds__(256) void mmd_finalize(const float* __restrict__ ws,
                                                    float* __restrict__ out) {
    __shared__ float sd[256];
    __shared__ float sw[256];
    const int tid = threadIdx.x;

    const float scale = unflipf(((const unsigned*)ws)[513]) -
                        unflipf(((const unsigned*)ws)[514]);
    const float inv = 1.0f / scale;

    float u = ws[tid] * inv;
    float v = ws[256 + tid] * inv;
    float d = u - v;
    float w = u + v;
    sd[tid] = d * d;
    sw[tid] = w * w;
    __syncthreads();
    for (int off = 128; off; off >>= 1) {
        if (tid < off) {
            sd[tid] += sd[tid + off];
            sw[tid] += sw[tid + off];
        }
        __syncthreads();
    }

    if (tid == 0) {
        const float diff2 = sd[0];          // ||u - v||^2
        const float sumw2 = sw[0];          // ||u + v||^2
        const float sumsq = ws[512] * inv * inv;  // sum over total of x^2
        const float nn = 2.0f * (float)BROWS;     // n = 8192

        // sum(L2) = (2n*sum(sq) - 2*||sum(total)||^2) / D
        const float sumL2 = (2.0f * nn * sumsq - 2.0f * sumw2) / (float)DDIM;
        float bw = sumL2 / (nn * nn - nn);
        bw *= 0.25f;                        // / KERNEL_MUL^(KERNEL_NUM//2) = 2^2

        float ibs = 0.0f;
        float bwi = bw;
#pragma unroll
        for (int i = 0; i < 5; ++i) {       // sum 1/(bw * 2^i)
            ibs += 1.0f / bwi;
            bwi *= 2.0f;
        }

        // mean(XX+YY-XY-YX) = (2*ibs/D) * ||u-v||^2 / B^2
        out[0] = 2.0f * ibs * diff2 /
                 ((float)DDIM * (float)BROWS * (float)BROWS);
    }
}

extern "C" void kernel_launch(void* const* d_in, const int* in_sizes, int n_in,
                              void* d_out, int out_size, void* d_ws, size_t ws_size,
                              hipStream_t stream) {
    (void)in_sizes; (void)n_in; (void)out_size; (void)ws_size;
    const float* src = (const float*)d_in[0];
    const float* tgt = (const float*)d_in[1];
    float* ws  = (float*)d_ws;
    float* out = (float*)d_out;

    mmd_zero<<<1, 256, 0, stream>>>(ws);
    mmd_reduce<<<NUM_BLOCKS, 256, 0, stream>>>(src, tgt, ws);
    mmd_finalize<<<1, 256, 0, stream>>>(ws, out);
}